// MAD_31164282700114
// MI455X (gfx1250) — compile-verified
//
#include <hip/hip_runtime.h>
#include <math.h>

typedef __attribute__((ext_vector_type(2))) float v2f;
typedef __attribute__((ext_vector_type(8))) float v8f;

#define N_NODES   10000
#define IN_FEATS  500
#define NODE_FEATS 64
#define N_SAMPLES 8
#define N_EDGES   131072

// ---------------------------------------------------------------------------
// Kernel 1: fused projection GEMM  f_all = feats@Wf+bf,  g_all = feats@Wg+bg
// via V_WMMA_F32_16X16X4_F32 (full fp32 precision, K=500 accumulation).
// grid = (625 row-tiles, 2), block = 128 threads = 4 waves; wave w owns the
// 16x16 tile at output columns [16w, 16w+16). 10000 = 625*16, 500 = 125*4,
// 64 = 4*16 -> no remainder handling, EXEC is all-ones at every WMMA.
// ---------------------------------------------------------------------------
__global__ __launch_bounds__(128)
void proj_gemm_wmma(const float* __restrict__ feats,
                    const float* __restrict__ Wf, const float* __restrict__ bf,
                    const float* __restrict__ Wg, const float* __restrict__ bg,
                    float* __restrict__ f_all, float* __restrict__ g_all)
{
    const float* W    = (blockIdx.y == 0) ? Wf : Wg;
    const float* bias = (blockIdx.y == 0) ? bf : bg;
    float*       outp = (blockIdx.y == 0) ? f_all : g_all;

    const int wave = threadIdx.x >> 5;
    const int lane = threadIdx.x & 31;
    const int half = lane >> 4;   // 0: holds K offsets {0,1}; 1: K offsets {2,3}
    const int lr   = lane & 15;

    const int m0 = blockIdx.x * 16;   // row tile base (nodes)
    const int n0 = wave * 16;         // col tile base (output features)

    // A fragment source: A[m][k] = feats[m*IN_FEATS + k], m = m0+lr, k = k0 + 2*half + r
    const float* arow = feats + (size_t)(m0 + lr) * IN_FEATS + 2 * half;
    // B fragment source: B[k][n] = W[k*NODE_FEATS + n], n = n0+lr, k = k0 + 2*half + r
    const float* bcol = W + (2 * half) * NODE_FEATS + n0 + lr;

    v8f acc = {};
    for (int k0 = 0; k0 < IN_FEATS; k0 += 4) {
        // contiguous pair -> single b64 load (even float offset => 8B aligned)
        float2 ap = *(const float2*)(arow + k0);
        v2f a; a.x = ap.x; a.y = ap.y;
        v2f b;
        b.x = bcol[(size_t)k0 * NODE_FEATS];
        b.y = bcol[(size_t)k0 * NODE_FEATS + NODE_FEATS];
        acc = __builtin_amdgcn_wmma_f32_16x16x4_f32(
            /*neg_a=*/false, a, /*neg_b=*/false, b,
            /*c_mod=*/(short)0, acc, /*reuse_a=*/false, /*reuse_b=*/false);
    }

    // C/D layout: lane l, reg r -> row M = r + 8*(l>=16), col N = l%16
    const float bv = bias[n0 + lr];
#pragma unroll
    for (int r = 0; r < 8; ++r) {
        outp[(size_t)(m0 + r + 8 * half) * NODE_FEATS + n0 + lr] = acc[r] + bv;
    }
}

// ---------------------------------------------------------------------------
// Kernel 2: one wave32 per edge. Lane layout: s = lane%8 (sample),
// q = lane/8 (feature quarter, 16 feats each). Partial dots/norms per lane,
// xor-8/16 reduces over q, width-8 butterflies handle the softmax over the
// 24 (d,logit) entries. Group-of-8 replicas are all valid; lane 0 stores.
// ---------------------------------------------------------------------------
__global__ __launch_bounds__(256)
void edge_kernel(const int* __restrict__ src, const int* __restrict__ dst,
                 const int* __restrict__ mid0, const int* __restrict__ mid1,
                 const float* __restrict__ mem,
                 const float* __restrict__ f_all, const float* __restrict__ g_all,
                 const float* __restrict__ adapt_w, const float* __restrict__ adapt_b,
                 float* __restrict__ out)
{
    const int e    = (int)((blockIdx.x * (unsigned)blockDim.x + threadIdx.x) >> 5);
    const int lane = threadIdx.x & 31;
    const int s    = lane & 7;
    const int q    = lane >> 3;

    const int sn = src[e];
    const int dn = dst[e];
    const int m0 = mid0[(size_t)e * N_SAMPLES + s];
    const int m1 = mid1[(size_t)e * N_SAMPLES + s];

    const float4* fs = (const float4*)(f_all + (size_t)sn * NODE_FEATS + q * 16);
    const float4* fd = (const float4*)(f_all + (size_t)dn * NODE_FEATS + q * 16);
    const float4* gs = (const float4*)(g_all + (size_t)sn * NODE_FEATS + q * 16);
    const float4* gd = (const float4*)(g_all + (size_t)dn * NODE_FEATS + q * 16);
    const float4* p0 = (const float4*)(f_all + (size_t)m0 * NODE_FEATS + q * 16);
    const float4* p1 = (const float4*)(f_all + (size_t)m1 * NODE_FEATS + q * 16);

    float dot1 = 0.f, nr1 = 0.f, dot2 = 0.f, nr2 = 0.f;
#pragma unroll
    for (int i = 0; i < 4; ++i) {
        float4 a = fs[i], b = p0[i], g = gd[i];
        float4 c = fd[i], d = p1[i], h = gs[i];
        float t;
        t = a.x - b.x; dot1 += t * g.x; nr1 += t * t;
        t = a.y - b.y; dot1 += t * g.y; nr1 += t * t;
        t = a.z - b.z; dot1 += t * g.z; nr1 += t * t;
        t = a.w - b.w; dot1 += t * g.w; nr1 += t * t;
        t = c.x - d.x; dot2 += t * h.x; nr2 += t * t;
        t = c.y - d.y; dot2 += t * h.y; nr2 += t * t;
        t = c.z - d.z; dot2 += t * h.z; nr2 += t * t;
        t = c.w - d.w; dot2 += t * h.w; nr2 += t * t;
    }

    // reduce over the 4 feature-quarter lanes sharing this sample s
    dot1 += __shfl_xor(dot1, 8, 32);  dot1 += __shfl_xor(dot1, 16, 32);
    nr1  += __shfl_xor(nr1,  8, 32);  nr1  += __shfl_xor(nr1,  16, 32);
    dot2 += __shfl_xor(dot2, 8, 32);  dot2 += __shfl_xor(dot2, 16, 32);
    nr2  += __shfl_xor(nr2,  8, 32);  nr2  += __shfl_xor(nr2,  16, 32);

    const float aw = adapt_w[0];
    const float ab = adapt_b[0];
    const float mm0 = mem[(size_t)m0 * N_NODES + dn];   // mem[mid0[e,s], dst[e]]
    const float mm1 = mem[(size_t)sn * N_NODES + m1];   // mem[src[e], mid1[e,s]]

    const float d1 = sqrtf(nr1);
    const float d2 = sqrtf(nr2);
    const float l1 = dot1 + aw * mm0 + ab;
    const float l2 = dot2 + aw * mm1 + ab;

    // softmax max over -dist  (dist includes the S=8 ones block)
    float dmin = fminf(d1, d2);
    dmin = fminf(dmin, __shfl_xor(dmin, 1, 32));
    dmin = fminf(dmin, __shfl_xor(dmin, 2, 32));
    dmin = fminf(dmin, __shfl_xor(dmin, 4, 32));
    dmin = fminf(dmin, 1.0f);

    const float e1 = expf(dmin - d1);     // exp(-d1 - max(-dist))
    const float e2 = expf(dmin - d2);
    float Z   = e1 + e2;
    float num = l1 * e1 + l2 * e2;
    Z   += __shfl_xor(Z, 1, 32);   Z   += __shfl_xor(Z, 2, 32);   Z   += __shfl_xor(Z, 4, 32);
    num += __shfl_xor(num, 1, 32); num += __shfl_xor(num, 2, 32); num += __shfl_xor(num, 4, 32);
    Z += 8.0f * expf(dmin - 1.0f);        // zero-logit / unit-dist block

    if (lane == 0) {
        const float r = num / Z;
        out[e] = 1.0f / (1.0f + expf(-r));
    }
}

// ---------------------------------------------------------------------------
extern "C" void kernel_launch(void* const* d_in, const int* in_sizes, int n_in,
                              void* d_out, int out_size, void* d_ws, size_t ws_size,
                              hipStream_t stream) {
    const int*   src   = (const int*)d_in[0];
    const int*   dst   = (const int*)d_in[1];
    const int*   mid0  = (const int*)d_in[2];
    const int*   mid1  = (const int*)d_in[3];
    const float* feats = (const float*)d_in[4];
    const float* mem   = (const float*)d_in[5];
    const float* Wf    = (const float*)d_in[6];
    const float* bf    = (const float*)d_in[7];
    const float* Wg    = (const float*)d_in[8];
    const float* bg    = (const float*)d_in[9];
    const float* aw    = (const float*)d_in[10];
    const float* ab    = (const float*)d_in[11];
    float*       out   = (float*)d_out;

    float* f_all = (float*)d_ws;                              // 10000*64 f32
    float* g_all = f_all + (size_t)N_NODES * NODE_FEATS;      // 10000*64 f32

    dim3 g1(N_NODES / 16, 2);
    proj_gemm_wmma<<<g1, 128, 0, stream>>>(feats, Wf, bf, Wg, bg, f_all, g_all);

    const int waves_per_block = 256 / 32;
    dim3 g2(N_EDGES / waves_per_block);
    edge_kernel<<<g2, 256, 0, stream>>>(src, dst, mid0, mid1, mem,
                                        f_all, g_all, aw, ab, out);
}